// TwoStepSpodNet_7095285973285
// MI455X (gfx1250) — compile-verified
//
#include <hip/hip_runtime.h>
#include <stdint.h>

typedef float v2f __attribute__((ext_vector_type(2)));
typedef float v8f __attribute__((ext_vector_type(8)));
typedef unsigned int u32x4 __attribute__((ext_vector_type(4)));
typedef int i32x4 __attribute__((ext_vector_type(4)));
typedef int i32x8 __attribute__((ext_vector_type(8)));

#define P   128
#define PM1 127
#define NT  256

// ---------------------------------------------------------------------------
// TDM: async DMA of a 128x128 f32 tile (row stride 128) global -> LDS.
// D# built per CDNA5 ISA. A plain copy after the wait
// guarantees correctness even if descriptor semantics differ from the probe.
// ---------------------------------------------------------------------------
__device__ __forceinline__ void tdm_load_tile_128x128_f32(const float* gsrc,
                                                          void* lds_dst) {
#if __has_builtin(__builtin_amdgcn_tensor_load_to_lds)
    const unsigned long long ga = (unsigned long long)(uintptr_t)gsrc;
    const unsigned int lds = (unsigned int)(uintptr_t)lds_dst; // LDS byte offset

    u32x4 g0;
    g0.x = 1u;                                   // count=1 (valid user D#)
    g0.y = lds;                                  // lds_addr
    g0.z = (unsigned int)(ga & 0xFFFFFFFFu);     // global_addr[31:0]
    g0.w = (unsigned int)((ga >> 32) & 0x01FFFFFFu) | (2u << 30); // [56:32]|type=2

    i32x8 g1;
    g1[0] = (int)(2u << 16);      // workgroup_mask=0, data_size=2 (4 bytes)
    g1[1] = (int)(128u << 16);    // tensor_dim0 = 128
    g1[2] = (int)(128u << 16);    // tensor_dim0 hi=0, tensor_dim1 = 128
    g1[3] = (int)(128u << 16);    // tensor_dim1 hi=0, tile_dim0 = 128
    g1[4] = 128;                  // tile_dim1 = 128, tile_dim2 = 0
    g1[5] = 128;                  // tensor_dim0_stride = 128
    g1[6] = 0;
    g1[7] = 0;

    i32x4 gz = {0, 0, 0, 0};
#if defined(__clang_major__) && (__clang_major__ >= 23)
    i32x8 gz8 = {0, 0, 0, 0, 0, 0, 0, 0};
    __builtin_amdgcn_tensor_load_to_lds(g0, g1, gz, gz, gz8, 0);
#else
    __builtin_amdgcn_tensor_load_to_lds(g0, g1, gz, gz, 0);
#endif
    __builtin_amdgcn_s_wait_tensorcnt(0);
#else
    (void)gsrc; (void)lds_dst;
#endif
}

// ---------------------------------------------------------------------------
// Kernel 1: per-batch  D = T12p @ L_w^T  via V_WMMA_F32_16X16X4_F32,
// then Theta_new[r][s] = (r==s) ? Theta[r][r] : D[max(r,s)][min(r,s)].
// One workgroup (8 waves) per batch element. L_w staged zero-padded in LDS.
// ---------------------------------------------------------------------------
__global__ __launch_bounds__(NT) void spod_delta_kernel(
    const float* __restrict__ theta, const float* __restrict__ Lw,
    float* __restrict__ out)
{
    __shared__ float sT[P * P];   // phase 1: T12^T (sT[k*P+c]); phase 2: D[c][j]
    __shared__ float sL[P * P];   // zero-padded L_w, row-major [j][k]

    const int b   = blockIdx.x;
    const int tid = threadIdx.x;
    const float* Tb = theta + (size_t)b * P * P;
    float*       Ob = out   + (size_t)b * P * P;

    // Stage L_w (zero-padded) and build T12^T: sT[k*P+c] = Theta[k+(k>=c)][c].
    if (tid < P) sT[PM1 * P + tid] = 0.0f;       // pad row k=127
    for (int idx = tid; idx < P * P; idx += NT) {
        int r = idx >> 7, c = idx & (P - 1);
        sL[idx] = (r < PM1 && c < PM1) ? Lw[r * PM1 + c] : 0.0f;
        float val = Tb[idx];                     // coalesced over c
        if (r != c) {
            int k = r - (r > c ? 1 : 0);
            sT[k * P + c] = val;
        }
    }
    __syncthreads();

    const int lane   = tid & 31;
    const int wv     = tid >> 5;      // this wave's N-tile (j-tile)
    const int laneM  = lane & 15;
    const int laneHi = lane >> 4;     // selects K pair {0,1} vs {2,3}
    const int jg     = wv * 16 + laneM;

    v8f acc[8] = {};                  // 8 M-tiles (c-tiles) x one N-tile per wave

    for (int k0 = 0; k0 < P; k0 += 4) {
        const int k = k0 + 2 * laneHi;
        // B[k][j] = L_w[j][k]; contiguous pair -> single aligned 8B LDS load
        const v2f bfrag = *(const v2f*)(sL + jg * P + k);
        #pragma unroll
        for (int ct = 0; ct < 8; ++ct) {
            const int m = ct * 16 + laneM;        // A row (c index)
            v2f afrag;
            afrag.x = sT[k       * P + m];        // A[m][k]
            afrag.y = sT[(k + 1) * P + m];        // A[m][k+1]
            acc[ct] = __builtin_amdgcn_wmma_f32_16x16x4_f32(
                false, afrag, false, bfrag, (short)0, acc[ct], false, false);
        }
    }
    __syncthreads();                  // all waves done reading sT (A operands)

    // Store D tiles into sT reused as sD[c][j] (C/D layout: VGPR v, lane-half -> M)
    #pragma unroll
    for (int ct = 0; ct < 8; ++ct) {
        #pragma unroll
        for (int v = 0; v < 8; ++v) {
            const int row = ct * 16 + v + 8 * laneHi;   // c
            const int col = wv * 16 + laneM;            // j
            sT[row * P + col] = acc[ct][v];
        }
    }
    __syncthreads();

    // Theta_new (diag = old diag; kernel 2 rewrites it)
    for (int idx = tid; idx < P * P; idx += NT) {
        int r = idx >> 7, s = idx & (P - 1);
        float val;
        if (r == s) {
            val = Tb[r * P + r];
        } else {
            int mx = r > s ? r : s;
            int mn = r > s ? s : r;
            val = sT[mx * P + mn];
        }
        Ob[idx] = val;
    }
}

// ---------------------------------------------------------------------------
// Kernel 2: per-batch W = inv(Theta_old) (Gauss-Jordan, SPD -> no pivoting),
// then the sequential 128-column sweep; writes only the diagonal of out.
// W stays symmetric -> column accesses done as row accesses (no bank conflicts).
// ---------------------------------------------------------------------------
__global__ __launch_bounds__(NT) void spod_sweep_kernel(
    const float* __restrict__ theta, float* __restrict__ out)
{
    __shared__ float sW[P * P];                         // 64 KB, W resident
    __shared__ float srow[P], scol[P];                  // GJ scratch
    __shared__ float t12v[P], d12v[P], uv[P], vv[P], wc[P];

    const int b   = blockIdx.x;
    const int tid = threadIdx.x;
    const float* Tb = theta + (size_t)b * P * P;
    float*       Ob = out   + (size_t)b * P * P;

    // Async TDM prefetch of Theta[b] into sW (wave 0 issues, waits TENSORcnt)
    if (tid < 32) tdm_load_tile_128x128_f32(Tb, sW);
    __syncthreads();
    // Backstop copy (same data; keeps kernel correct independent of D# semantics)
    for (int idx = tid; idx < P * P; idx += NT) sW[idx] = Tb[idx];
    __syncthreads();

    // ---- In-place Gauss-Jordan inversion ----
    for (int k = 0; k < P; ++k) {
        const float invp = 1.0f / sW[k * P + k];
        if (tid < P) {
            srow[tid] = ((tid == k) ? 1.0f : sW[k * P + tid]) * invp;
            scol[tid] = (tid == k) ? 0.0f : sW[tid * P + k];
        }
        __syncthreads();
        for (int idx = tid; idx < P * P; idx += NT) {
            int i = idx >> 7, j = idx & (P - 1);
            if (i == k) {
                sW[idx] = srow[j];
            } else {
                float a = (j == k) ? 0.0f : sW[idx];
                sW[idx] = a - scol[i] * srow[j];
            }
        }
        __syncthreads();
    }

    // ---- Sequential column sweep ----
    for (int c = 0; c < P; ++c) {
        __syncthreads();
        if (c + 1 < P)
            __builtin_prefetch(&Ob[(tid & (P - 1)) * P + c + 1], 0, 1);
        if (tid < P) {
            float tn = Ob[tid * P + c];          // Theta_new column c
            float to = Tb[tid * P + c];          // Theta_old column c
            if (tid == c) { tn = 0.0f; to = 0.0f; }
            t12v[tid] = tn;
            d12v[tid] = tn - to;                 // Delta column c
        }
        __syncthreads();

        // raw matvecs: uv = W*d12, vv = W*t12 (one row per thread-half)
        if (tid < P) {
            const float* wr = sW + tid * P;
            float a = 0.0f;
            for (int s = 0; s < P; ++s) a += wr[s] * d12v[s];
            uv[tid] = a;
        } else {
            const int r = tid - P;
            const float* wr = sW + r * P;
            float a = 0.0f;
            for (int s = 0; s < P; ++s) a += wr[s] * t12v[s];
            vv[r] = a;
        }
        __syncthreads();

        const float w22 = sW[c * P + c];
        const float au  = uv[c] / w22;
        const float av  = vv[c] / w22;
        float ur = 0.0f, vr = 0.0f;
        if (tid < P && tid != c) {
            const float wrc = sW[c * P + tid];   // W[tid][c] via symmetry
            ur = uv[tid] - wrc * au;             // (inv_T11 * d12)[tid]
            vr = vv[tid] - wrc * av;             // (inv_T11 * t12)[tid]
        }
        __syncthreads();
        if (tid < P) { uv[tid] = ur; vv[tid] = vr; wc[tid] = sW[c * P + tid]; }
        __syncthreads();

        // scalars (computed redundantly, identically, by every thread)
        float schur = 0.0f, tdot = 0.0f;
        for (int s = 0; s < P; ++s) {
            schur += d12v[s] * uv[s];
            tdot  += t12v[s] * vv[s];
        }
        const float t22   = Tb[c * P + c];
        const float t22n  = t22 + schur;
        const float w22n  = 1.0f / (t22n - tdot);
        if (tid == 0) Ob[c * P + c] = t22n;      // new diagonal entry

        // symmetric rank-2 update of W; row/col c replaced
        const float invw22 = 1.0f / w22;
        for (int idx = tid; idx < P * P; idx += NT) {
            int i = idx >> 7, j = idx & (P - 1);
            float val;
            if (i == c) {
                val = (j == c) ? w22n : (-w22n * vv[j]);
            } else if (j == c) {
                val = -w22n * vv[i];
            } else {
                val = sW[idx] - wc[i] * wc[j] * invw22 + w22n * vv[i] * vv[j];
            }
            sW[idx] = val;
        }
    }
}

// ---------------------------------------------------------------------------
extern "C" void kernel_launch(void* const* d_in, const int* in_sizes, int n_in,
                              void* d_out, int out_size, void* d_ws, size_t ws_size,
                              hipStream_t stream) {
    const float* theta = (const float*)d_in[0];   // (128,128,128) f32
    const float* Lw    = (const float*)d_in[1];   // (127,127) f32
    float* out = (float*)d_out;                   // (128,128,128) f32
    (void)in_sizes; (void)n_in; (void)out_size; (void)d_ws; (void)ws_size;

    spod_delta_kernel<<<128, NT, 0, stream>>>(theta, Lw, out);
    spod_sweep_kernel<<<128, NT, 0, stream>>>(theta, out);
}